// residualconv_59330678227578
// MI455X (gfx1250) — compile-verified
//
#include <hip/hip_runtime.h>

typedef __attribute__((ext_vector_type(2))) float v2f;
typedef __attribute__((ext_vector_type(4))) float f32x4;
typedef __attribute__((ext_vector_type(8))) float v8f;
typedef __attribute__((ext_vector_type(4))) int   v4i;

#define ALPHA 0.05f

static constexpr int B = 8, C = 32, V = 512, L = 256;
static constexpr int NC = B * C;                      // 256 (n,c) planes
static constexpr size_t PLANE  = (size_t)V * L;       // 131072 elements
static constexpr size_t APLANE = (size_t)V * V;       // 262144 elements

// ---------------------------------------------------------------------------
// Async global->LDS staging (CDNA5 ASYNCcnt path), with synchronous fallback.
// ROCm 7.2 signature: (v4i AS1* src, v4i AS3* dst, imm int offset, imm int cpol)
// ---------------------------------------------------------------------------
#if __has_builtin(__builtin_amdgcn_global_load_async_to_lds_b128)
#define HAVE_ASYNC_LDS 1
typedef __attribute__((address_space(1))) v4i GV4;   // global int4
typedef __attribute__((address_space(3))) v4i LV4;   // LDS int4
#endif

__device__ __forceinline__ void stage16(const float* g, float* l)
{
#ifdef HAVE_ASYNC_LDS
    __builtin_amdgcn_global_load_async_to_lds_b128(
        (GV4*)g, (LV4*)l, /*offset=*/0, /*cpol=*/0);
#else
    *(f32x4*)l = *(const f32x4*)g;
#endif
}

__device__ __forceinline__ void wait_stage()
{
#ifdef HAVE_ASYNC_LDS
#if __has_builtin(__builtin_amdgcn_s_wait_asynccnt)
    __builtin_amdgcn_s_wait_asynccnt(0);
#else
    asm volatile("s_wait_asynccnt 0x0" ::: "memory");
#endif
#endif
}

// ---------------------------------------------------------------------------
// 1) a[n,v,w] = (adj[n,v,w] + (v==w)) / sum_v(adj[n,v,w] + (v==w))
//    (column-normalized adjacency; coalesced across w)
// ---------------------------------------------------------------------------
__global__ __launch_bounds__(256)
void norm_adj_kernel(const float* __restrict__ adj, float* __restrict__ anorm)
{
    const int n = blockIdx.y;
    const int w = blockIdx.x * 256 + threadIdx.x;
    const float* src = adj   + (size_t)n * APLANE;
    float*       dst = anorm + (size_t)n * APLANE;

    float d = 0.f;
    for (int v = 0; v < V; ++v)
        d += src[(size_t)v * V + w] + (v == w ? 1.f : 0.f);
    const float inv = 1.f / d;
    for (int v = 0; v < V; ++v)
        dst[(size_t)v * V + w] = (src[(size_t)v * V + w] + (v == w ? 1.f : 0.f)) * inv;
}

// ---------------------------------------------------------------------------
// 2) Batched WMMA GEMM, two flavors:
//    mode 0: O[nc] = A[nc](512xK) * Bm(Kx256) + bias[l]           (hw = h*W + b)
//    mode 1: O[nc] = ALPHA*X[nc] + (1-ALPHA) * A[n](512xK)*Bm[nc] (h = res + a*hw)
//
//    Block: 256 threads = 8 waves; block tile 64(M) x 128(N).
//    Wave tile 32x32 => 2x2 accumulators of v_wmma_f32_16x16x4_f32.
//    A double-buffered in LDS via async global->LDS b128 copies (ASYNCcnt),
//    K-chunks of 128, row stride 132 => conflict-free b64 fragment reads.
//    B fragments read from global (W / hw stay hot in the 192MB L2).
// ---------------------------------------------------------------------------
static constexpr int KTILE = 128;
static constexpr int LDSP  = KTILE + 4;   // padded LDS row stride (floats)

__device__ __forceinline__ void stage_chunk(const float* __restrict__ A, int m0,
                                            int K, int kc, float* __restrict__ buf,
                                            int tid)
{
    const float* src = A + (size_t)m0 * K + (size_t)kc * KTILE;
    #pragma unroll
    for (int i = 0; i < 8; ++i) {
        int idx = tid + i * 256;      // 2048 float4 units total
        int row = idx >> 5;           // 32 float4 per row
        int c4  = idx & 31;
        stage16(src + (size_t)row * K + c4 * 4, buf + row * LDSP + c4 * 4);
    }
}

__global__ __launch_bounds__(256, 2)
void gemm_kernel(const float* __restrict__ Abase,  // planes of (512 x K), row-major
                 const float* __restrict__ Bbase,  // (K x 256) [+ optional plane stride]
                 const float* __restrict__ bias,   // L floats (mode 0)
                 const float* __restrict__ xbase,  // residual planes (mode 1)
                 float* __restrict__ obase,        // output planes (nc, 512, 256)
                 int K, int per_channel_A, size_t b_plane_stride, int mode)
{
    __shared__ float Alds[2][64 * LDSP];

    const int nc = blockIdx.z;
    const int m0 = blockIdx.x * 64;
    const int n0 = blockIdx.y * 128;

    const int plane_a = per_channel_A ? nc : (nc >> 5);   // nc or n
    const float* A  = Abase + (size_t)plane_a * (size_t)V * (size_t)K;
    const float* Bm = Bbase + (size_t)nc * b_plane_stride;
    float*       O  = obase + (size_t)nc * PLANE;
    const float* X  = xbase ? xbase + (size_t)nc * PLANE : nullptr;

    const int tid  = threadIdx.x;
    const int lane = tid & 31;
    const int wave = tid >> 5;
    const int wm   = wave >> 2;      // 0..1  (M)
    const int wn   = wave & 3;       // 0..3  (N)
    const int g    = lane >> 4;      // 0..1  (K-pair group / M-half)
    const int r    = lane & 15;      // row/col within 16

    v8f acc[2][2];
    {
        v8f z = {};
        acc[0][0] = z; acc[0][1] = z; acc[1][0] = z; acc[1][1] = z;
    }

    const int nchunks = K / KTILE;

    // prologue: stage chunk 0
    stage_chunk(A, m0, K, 0, &Alds[0][0], tid);
    wait_stage();
    __syncthreads();

    for (int kc = 0; kc < nchunks; ++kc) {
        const float* __restrict__ Acur = &Alds[kc & 1][0];

        // overlap: kick off the next chunk's async copy into the other buffer
        // (that buffer was last *read* before the previous barrier -> no WAR hazard)
        if (kc + 1 < nchunks)
            stage_chunk(A, m0, K, kc + 1, &Alds[(kc + 1) & 1][0], tid);

        const int kglob = kc * KTILE;
        #pragma unroll 4
        for (int kk = 0; kk < KTILE; kk += 4) {
            // A fragments: element (m,k): lane = m + 16*(k>>1), vgpr = k&1
            v2f afrag[2];
            #pragma unroll
            for (int mt = 0; mt < 2; ++mt) {
                int row = wm * 32 + mt * 16 + r;
                afrag[mt] = *(const v2f*)(&Acur[row * LDSP + kk + 2 * g]);
            }
            // B fragments: element (k,n): lane = n + 16*(k>>1), vgpr = k&1
            v2f bfrag[2];
            #pragma unroll
            for (int nt = 0; nt < 2; ++nt) {
                int col  = n0 + wn * 32 + nt * 16 + r;
                int krow = kglob + kk + 2 * g;
                v2f bf;
                bf.x = Bm[(size_t)krow * L + col];
                bf.y = Bm[(size_t)(krow + 1) * L + col];
                bfrag[nt] = bf;
            }
            #pragma unroll
            for (int mt = 0; mt < 2; ++mt)
                #pragma unroll
                for (int nt = 0; nt < 2; ++nt)
                    acc[mt][nt] = __builtin_amdgcn_wmma_f32_16x16x4_f32(
                        false, afrag[mt], false, bfrag[nt],
                        (short)0, acc[mt][nt], false, false);
        }

        if (kc + 1 < nchunks) {
            wait_stage();      // this wave's async copies landed in LDS
            __syncthreads();   // everyone's copies landed; safe to swap buffers
        }
    }

    // Epilogue. D element (m,n): lane = n + 16*(m>=8), vgpr i -> m = i + 8*(lane>>4)
    #pragma unroll
    for (int mt = 0; mt < 2; ++mt) {
        #pragma unroll
        for (int nt = 0; nt < 2; ++nt) {
            const int col  = n0 + wn * 32 + nt * 16 + r;
            const float bv = (mode == 0) ? bias[col] : 0.f;
            #pragma unroll
            for (int i = 0; i < 8; ++i) {
                const int row  = m0 + wm * 32 + mt * 16 + 8 * g + i;
                const size_t off = (size_t)row * L + col;
                const float v = acc[mt][nt][i];
                if (mode == 0) {
                    O[off] = v + bv;
                } else {
                    O[off] = ALPHA * X[off] + (1.f - ALPHA) * v;
                }
            }
        }
    }
}

// ---------------------------------------------------------------------------
// 3) out[n,o,v,l] = sum_{c<96} w2[o,c] * cat(x,h1,h2)[n,c,v,l] + conv_b[o]
//    Memory-bound channel mix; weights in LDS, 32 accumulators per thread.
// ---------------------------------------------------------------------------
__global__ __launch_bounds__(256)
void conv_out_kernel(const float* __restrict__ x,  const float* __restrict__ h1,
                     const float* __restrict__ h2, const float* __restrict__ cw,
                     const float* __restrict__ cb, float* __restrict__ out)
{
    __shared__ float sw[96 * 32];   // sw[c*32 + o] = conv_w[o, c]
    __shared__ float sb[32];
    for (int i = threadIdx.x; i < 96 * 32; i += 256) {
        int o = i & 31, c = i >> 5;
        sw[i] = cw[o * 96 + c];
    }
    if (threadIdx.x < 32) sb[threadIdx.x] = cb[threadIdx.x];
    __syncthreads();

    const size_t tid = (size_t)blockIdx.x * 256 + threadIdx.x;   // over B * V*L
    const int    n   = (int)(tid / PLANE);
    const size_t p   = tid % PLANE;

    const float* bx = x  + ((size_t)n * C) * PLANE + p;
    const float* b1 = h1 + ((size_t)n * C) * PLANE + p;
    const float* b2 = h2 + ((size_t)n * C) * PLANE + p;

    float acc[32];
    #pragma unroll
    for (int o = 0; o < 32; ++o) acc[o] = sb[o];

    #pragma unroll 4
    for (int c = 0; c < 32; ++c) {
        float s = bx[(size_t)c * PLANE];
        const float* w = &sw[c * 32];
        #pragma unroll
        for (int o = 0; o < 32; ++o) acc[o] = fmaf(s, w[o], acc[o]);
    }
    #pragma unroll 4
    for (int c = 0; c < 32; ++c) {
        float s = b1[(size_t)c * PLANE];
        const float* w = &sw[(32 + c) * 32];
        #pragma unroll
        for (int o = 0; o < 32; ++o) acc[o] = fmaf(s, w[o], acc[o]);
    }
    #pragma unroll 4
    for (int c = 0; c < 32; ++c) {
        float s = b2[(size_t)c * PLANE];
        const float* w = &sw[(64 + c) * 32];
        #pragma unroll
        for (int o = 0; o < 32; ++o) acc[o] = fmaf(s, w[o], acc[o]);
    }

    float* bo = out + ((size_t)n * C) * PLANE + p;
    #pragma unroll
    for (int o = 0; o < 32; ++o)
        __builtin_nontemporal_store(acc[o], &bo[(size_t)o * PLANE]);
}

// ---------------------------------------------------------------------------
// Launcher.  Workspace layout (floats):
//   a  : 8*512*512          =   2,097,152   (8 MB)
//   hw : 256*512*256        =  33,554,432   (128 MB)
//   h1 : 33,554,432         (128 MB)
//   h2 : 33,554,432         (128 MB)
// ---------------------------------------------------------------------------
extern "C" void kernel_launch(void* const* d_in, const int* in_sizes, int n_in,
                              void* d_out, int out_size, void* d_ws, size_t ws_size,
                              hipStream_t stream)
{
    const float* x   = (const float*)d_in[0];
    const float* adj = (const float*)d_in[1];
    const float* W   = (const float*)d_in[2];
    const float* b   = (const float*)d_in[3];
    const float* cw  = (const float*)d_in[4];
    const float* cb  = (const float*)d_in[5];
    float* out = (float*)d_out;

    float* a  = (float*)d_ws;
    float* hw = a  + (size_t)B * APLANE;
    float* h1 = hw + (size_t)NC * PLANE;
    float* h2 = h1 + (size_t)NC * PLANE;

    // 1) normalized adjacency
    norm_adj_kernel<<<dim3(V / 256, B), 256, 0, stream>>>(adj, a);

    const dim3 ggrid(V / 64, L / 128, NC);
    // layer 1: hw = x*W + b ; h1 = a*hw residual-blended with x
    gemm_kernel<<<ggrid, 256, 0, stream>>>(x,  W,  b,       nullptr, hw, L, 1, 0,     0);
    gemm_kernel<<<ggrid, 256, 0, stream>>>(a,  hw, nullptr, x,       h1, V, 0, PLANE, 1);
    // layer 2: hw = h1*W + b ; h2 = a*hw residual-blended with x
    gemm_kernel<<<ggrid, 256, 0, stream>>>(h1, W,  b,       nullptr, hw, L, 1, 0,     0);
    gemm_kernel<<<ggrid, 256, 0, stream>>>(a,  hw, nullptr, x,       h2, V, 0, PLANE, 1);
    // 3) output 1x1 conv over [x, h1, h2]
    conv_out_kernel<<<dim3((unsigned)((B * PLANE) / 256)), 256, 0, stream>>>(
        x, h1, h2, cw, cb, out);
}